// GraphAttentionLayer_3350074490947
// MI455X (gfx1250) — compile-verified
//
#include <hip/hip_runtime.h>
#include <hip/hip_bf16.h>
#include <math.h>

// Problem constants (match reference)
#define BS   16
#define NN   256
#define DIN  128
#define HID  128
#define DOUT 128

typedef __attribute__((ext_vector_type(2))) float v2f;
typedef __attribute__((ext_vector_type(8))) float v8f;

// fp32 WMMA 16x16x4 — confirmed on gfx1250 (emits v_wmma_f32_16x16x4_f32).
// Guard keeps the host-compilation pass happy.
__device__ __forceinline__ v8f wmma_f32x4(v2f a, v2f b, v8f c) {
#if defined(__HIP_DEVICE_COMPILE__) && __has_builtin(__builtin_amdgcn_wmma_f32_16x16x4_f32)
  return __builtin_amdgcn_wmma_f32_16x16x4_f32(false, a, false, b, (short)0, c,
                                               false, false);
#else
  return c;
#endif
}

// Strided-batched GEMM, N fixed at 128, K compile-time (128 or 256):
//   C[b](Mx128) = A[b](MxK) * B[b](Kx128) (+ bias)
// Block = 128 threads (4 waves). Wave w owns a 16x32 output slab
// (n = w*32 .. w*32+31) -> 2 accumulators sharing one A fragment per k-step
// (A reuse x2, two independent WMMA dep-chains for the matrix pipe).
// grid = (M/16, batches).
template<int K>
__global__ void __launch_bounds__(128)
wmma_gemm_bias(const float* __restrict__ A, const float* __restrict__ B,
               const float* __restrict__ bias, float* __restrict__ C,
               long long sA, long long sB, long long sC)
{
  constexpr int N = 128;
  const int lane = threadIdx.x & 31;
  const int wave = threadIdx.x >> 5;
  const int half = lane >> 4;   // 0: lanes 0-15, 1: lanes 16-31
  const int lm   = lane & 15;

  const int m0 = blockIdx.x * 16;
  const int b  = blockIdx.y;
  const int n0 = wave * 32;

  const float* Ab = A + (long long)b * sA;
  const float* Bb = B + (long long)b * sB;
  float*       Cb = C + (long long)b * sC;

  // Streaming pointers: all per-iteration addressing folds into immediates.
  // A 16x4 frag (ISA 7.12.2): lane holds row m=lm, VGPR0/1 = K-pair (2*half, 2*half+1).
  // B 4x16 frag: VGPR r holds row (2*half + r), col = lm.
  const float* ap = Ab + (long long)(m0 + lm) * K + half * 2;
  const float* bp = Bb + (half * 2) * N + n0 + lm;

  v8f acc0 = {}, acc1 = {};
  #pragma unroll 8
  for (int k = 0; k < K; k += 4) {
    v2f a, b0, b1;
    a.x  = ap[0];          // consecutive pair -> b64 load
    a.y  = ap[1];
    b0.x = bp[0];
    b0.y = bp[N];
    b1.x = bp[16];
    b1.y = bp[N + 16];
    acc0 = wmma_f32x4(a, b0, acc0);
    acc1 = wmma_f32x4(a, b1, acc1);
    ap += 4;
    bp += 4 * N;
  }

  // C/D 16x16 f32 layout: VGPR r -> row (half*8 + r), col = lane%16.
  const int col0 = n0 + lm;
  const float bias0 = bias ? bias[col0]      : 0.f;
  const float bias1 = bias ? bias[col0 + 16] : 0.f;
  #pragma unroll
  for (int r = 0; r < 8; ++r) {
    const long long row = m0 + half * 8 + r;
    Cb[row * N + col0]      = acc0[r] + bias0;
    Cb[row * N + col0 + 16] = acc1[r] + bias1;
  }
}

// Pairwise relu-dot attention scores + masked softmax (irreducibly VALU:
// the ReLU sits inside the k-reduction, so this is not a GEMM).
// grid = BS * (N/16) blocks; block = 256 threads; block owns a 16-row i-tile.
__global__ void __launch_bounds__(256)
gat_scores_softmax(const float* __restrict__ ai, const float* __restrict__ aj,
                   const float* __restrict__ b_a1, const float* __restrict__ w_a2,
                   const float* __restrict__ b_a2, const int* __restrict__ adj,
                   float* __restrict__ attn)
{
  __shared__ float sAi[16][HID + 4];  // row stride 132 floats = 528 B:
  __shared__ float sAj[16][HID + 4];  // 16B-aligned for float4, conflict-free
  __shared__ float sB1[HID];
  __shared__ float sW2[HID];
  __shared__ float sE[16][NN];

  const int tid = threadIdx.x;
  const int b   = blockIdx.x >> 4;
  const int i0  = (blockIdx.x & 15) * 16;

  if (tid < HID) { sB1[tid] = b_a1[tid]; sW2[tid] = w_a2[tid]; }
  for (int idx = tid; idx < 16 * HID; idx += 256) {
    const int r = idx >> 7, c = idx & (HID - 1);
    sAi[r][c] = ai[((long long)(b * NN + i0 + r)) * HID + c];
  }
  const float ba2 = b_a2[0];
  const int i  = tid >> 4;   // local row 0..15
  const int jl = tid & 15;   // local col 0..15

  for (int jt = 0; jt < NN / 16; ++jt) {
    const int j0 = jt * 16;
    __syncthreads();  // prior compute done before overwriting sAj
    for (int idx = tid; idx < 16 * HID; idx += 256) {
      const int r = idx >> 7, c = idx & (HID - 1);
      sAj[r][c] = aj[((long long)(b * NN + j0 + r)) * HID + c];
    }
    __syncthreads();

    float acc = 0.f;
    #pragma unroll 4
    for (int k = 0; k < HID; k += 4) {
      const float4 vi = *(const float4*)&sAi[i][k];
      const float4 vj = *(const float4*)&sAj[jl][k];
      const float4 vb = *(const float4*)&sB1[k];
      const float4 vw = *(const float4*)&sW2[k];
      acc = fmaf(fmaxf(vi.x + vj.x + vb.x, 0.f), vw.x, acc);
      acc = fmaf(fmaxf(vi.y + vj.y + vb.y, 0.f), vw.y, acc);
      acc = fmaf(fmaxf(vi.z + vj.z + vb.z, 0.f), vw.z, acc);
      acc = fmaf(fmaxf(vi.w + vj.w + vb.w, 0.f), vw.w, acc);
    }
    const int jg = j0 + jl;
    const int ad = adj[((long long)(b * NN + i0 + i)) * NN + jg];
    sE[i][jg] = (ad == 0) ? -1e9f : (acc + ba2);
  }
  __syncthreads();

  // Row softmax: 16 threads per row, 16 strided cols each; 16-wide shfl_xor
  // reductions inside the wave32.
  float mx = -3.402823466e38f;
  for (int j = jl; j < NN; j += 16) mx = fmaxf(mx, sE[i][j]);
  #pragma unroll
  for (int m = 1; m < 16; m <<= 1) mx = fmaxf(mx, __shfl_xor(mx, m, 16));

  float sum = 0.f;
  for (int j = jl; j < NN; j += 16) sum += __expf(sE[i][j] - mx);
  #pragma unroll
  for (int m = 1; m < 16; m <<= 1) sum += __shfl_xor(sum, m, 16);

  const float inv = 1.f / sum;
  for (int j = jl; j < NN; j += 16)
    attn[((long long)(b * NN + i0 + i)) * NN + j] = __expf(sE[i][j] - mx) * inv;
}

extern "C" void kernel_launch(void* const* d_in, const int* in_sizes, int n_in,
                              void* d_out, int out_size, void* d_ws, size_t ws_size,
                              hipStream_t stream) {
  (void)in_sizes; (void)n_in; (void)out_size; (void)ws_size;
  const float* x     = (const float*)d_in[0];
  const int*   adj   = (const int*)  d_in[1];
  const float* W_fc  = (const float*)d_in[2];
  const float* b_fc  = (const float*)d_in[3];
  const float* W_a1  = (const float*)d_in[4];  // (2*HID, HID): rows 0..127 -> a_i, 128..255 -> a_j
  const float* b_a1  = (const float*)d_in[5];
  const float* w_a2  = (const float*)d_in[6];
  const float* b_a2  = (const float*)d_in[7];
  const float* W_out = (const float*)d_in[8];
  const float* b_out = (const float*)d_in[9];
  float* out = (float*)d_out;

  float* ws = (float*)d_ws;
  const long long HB = (long long)BS * NN * HID;          // 524288 floats
  float* h    = ws;
  float* ai   = ws + HB;
  float* aj   = ws + 2 * HB;
  float* attn = ws + 3 * HB;                              // BS*N*N floats
  float* hp   = ws + 3 * HB + (long long)BS * NN * NN;    // total ~12.6 MB

  const dim3 blk(128);
  const dim3 g(NN / 16, BS);   // (16, 16): one 16-row tile per block, 4 waves x 16x32

  // 1) h = x @ W_fc + b_fc        (B shared across batch: sB = 0)
  wmma_gemm_bias<DIN><<<g, blk, 0, stream>>>(x, W_fc, b_fc, h,
      (long long)NN * DIN, 0, (long long)NN * HID);
  // 2) a_i = h @ W_a1[:HID]
  wmma_gemm_bias<HID><<<g, blk, 0, stream>>>(h, W_a1, nullptr, ai,
      (long long)NN * HID, 0, (long long)NN * HID);
  // 3) a_j = h @ W_a1[HID:]
  wmma_gemm_bias<HID><<<g, blk, 0, stream>>>(h, W_a1 + (long long)HID * HID,
      nullptr, aj,
      (long long)NN * HID, 0, (long long)NN * HID);
  // 4) e/mask/softmax -> attn
  gat_scores_softmax<<<dim3(BS * (NN / 16)), dim3(256), 0, stream>>>(
      ai, aj, b_a1, w_a2, b_a2, adj, attn);
  // 5) h' = attn @ h              (both operands batched, K = 256)
  wmma_gemm_bias<NN><<<g, blk, 0, stream>>>(attn, h, nullptr, hp,
      (long long)NN * NN, (long long)NN * HID, (long long)NN * HID);
  // 6) out = h' @ W_out + b_out
  wmma_gemm_bias<HID><<<g, blk, 0, stream>>>(hp, W_out, b_out, out,
      (long long)NN * HID, 0, (long long)NN * DOUT);
}